// NXROAttentiveModel_10127532884087
// MI455X (gfx1250) — compile-verified
//
#include <hip/hip_runtime.h>
#include <math.h>

typedef __attribute__((ext_vector_type(2))) float v2f;
typedef __attribute__((ext_vector_type(8))) float v8f;

#define WPB 8  // waves per block

// out[b,i] = dxdt[b,i] + c2 * softmax3(x)_i * sigmoid(emb.aw)
// dxdt via fp32 WMMA GEMM: A[b, 16k+v] = emb[b,k]*x[b,v], W2[16k+v, u] = L_basis[k,u,v]
__global__ __launch_bounds__(256) void nxro_fused_kernel(
    const float* __restrict__ x, const float* __restrict__ t,
    const float* __restrict__ Lb, const float* __restrict__ wq,
    const float* __restrict__ wk, const float* __restrict__ wv,
    const float* __restrict__ wo, const float* __restrict__ aw,
    float* __restrict__ out, int nTiles)
{
    __shared__ float xs[WPB][16][18];  // x tile; row stride 72B (8B aligned, bank-skewed)
    __shared__ float al[WPB][16];      // alpha per sample

    const int lane = threadIdx.x & 31;
    const int wave = threadIdx.x >> 5;
    const int half = lane >> 4;   // 0: K-offsets {0,1}; 1: K-offsets {2,3}
    const int r    = lane & 15;   // A: sample row; B/D: column index

    // ---- constant B fragments for all 20 WMMA K-steps ----
    // B layout (16x16x4 f32): VGPR0: lanes0-15 K=4j+0, lanes16-31 K=4j+2;
    //                         VGPR1: lanes0-15 K=4j+1, lanes16-31 K=4j+3 (N = lane&15)
    v2f bfr[20];
#pragma unroll
    for (int j = 0; j < 20; ++j) {
        const int kb = j >> 2;                       // basis index
        const int v0 = ((j & 3) << 2) + (half << 1); // first var index of this K-pair
        const float* p = Lb + kb * 256 + r * 16 + v0;
        v2f b_; b_.x = p[0]; b_.y = p[1];
        bfr[j] = b_;
    }

    // ---- collapsed attention scalars ----
    float c1 = 0.f, c2 = 0.f;
#pragma unroll
    for (int d = 0; d < 32; ++d) {
        c1 = fmaf(wq[d], wk[d], c1);
        c2 = fmaf(wv[d], wo[d], c2);
    }
    c1 *= 0.17677669529663688f;  // 1/sqrt(32)

    const float aw0 = aw[0], aw1 = aw[1], aw2 = aw[2], aw3 = aw[3], aw4 = aw[4];

    const int waveId = blockIdx.x * WPB + wave;
    const int stride = gridDim.x * WPB;

    for (int tile = waveId; tile < nTiles; tile += stride) {
        const int s0  = tile << 4;
        const int row = s0 + r;   // sample this lane owns for the A operand

        // Each lane loads only the 4 x-pairs its K-half consumes: cols 4g + 2*half
        const v2f* xp = reinterpret_cast<const v2f*>(x + row * 16);
        const v2f xq0 = xp[0 + half];
        const v2f xq1 = xp[2 + half];
        const v2f xq2 = xp[4 + half];
        const v2f xq3 = xp[6 + half];

        // Cooperative x-tile stage: the two halves cover the full row
        {
            v2f* rp = reinterpret_cast<v2f*>(&xs[wave][r][0]);
            rp[0 + half] = xq0;
            rp[2 + half] = xq1;
            rp[4 + half] = xq2;
            rp[6 + half] = xq3;
        }

        // Fourier embedding. fl(4pi) == 2*fl(2pi) exactly, so the second
        // harmonic's argument equals 2*a1 bit-exactly; use double-angle
        // identities instead of a second sin/cos pair (saves ~100 VALU).
        const float tt  = t[row];
        const float a1  = 6.283185307179586f * tt;
        const float e1c = cosf(a1), e1s = sinf(a1);
        const float tc  = e1c + e1c;                 // 2*cos(a1)
        const float e2c = fmaf(tc, e1c, -1.0f);      // cos(2*a1)
        const float e2s = tc * e1s;                  // sin(2*a1)

        if (half == 0) {
            float z = aw0 + aw1 * e1c + aw2 * e1s + aw3 * e2c + aw4 * e2s;
            al[wave][r] = __builtin_amdgcn_rcpf(1.f + __expf(-z));
        }

        // ---- dxdt = A * W2 via 20 chained fp32 WMMA (fully explicit, no dyn idx) ----
        v8f acc = {};
#define WMMA_RAW(AFR, BJ)                                                      \
        acc = __builtin_amdgcn_wmma_f32_16x16x4_f32(                           \
            false, (AFR), false, bfr[BJ], (short)0, acc, false, false)
#define WMMA_SC(EV, XQ, BJ)                                                    \
        do { v2f a_; a_.x = (EV) * (XQ).x; a_.y = (EV) * (XQ).y;               \
             WMMA_RAW(a_, BJ); } while (0)

        WMMA_RAW(xq0, 0);  WMMA_RAW(xq1, 1);  WMMA_RAW(xq2, 2);  WMMA_RAW(xq3, 3);
        WMMA_SC(e1c, xq0, 4);  WMMA_SC(e1c, xq1, 5);  WMMA_SC(e1c, xq2, 6);  WMMA_SC(e1c, xq3, 7);
        WMMA_SC(e1s, xq0, 8);  WMMA_SC(e1s, xq1, 9);  WMMA_SC(e1s, xq2, 10); WMMA_SC(e1s, xq3, 11);
        WMMA_SC(e2c, xq0, 12); WMMA_SC(e2c, xq1, 13); WMMA_SC(e2c, xq2, 14); WMMA_SC(e2c, xq3, 15);
        WMMA_SC(e2s, xq0, 16); WMMA_SC(e2s, xq1, 17); WMMA_SC(e2s, xq2, 18); WMMA_SC(e2s, xq3, 19);
#undef WMMA_SC
#undef WMMA_RAW

        // ---- epilogue: masked 3-term softmax + store ----
        // D layout: VGPR rr holds (M = rr + 8*half, N = lane&15)
        const int i = r;
        const float g = (i >= 2) ? 1.f : 0.f;  // diag term distinct only for i>=2
#pragma unroll
        for (int rr = 0; rr < 8; ++rr) {
            const int m = rr + (half << 3);
            const v2f xm01 = *reinterpret_cast<const v2f*>(&xs[wave][m][0]);
            const float xm0 = xm01.x;
            const float xm1 = xm01.y;
            const float xmi = xs[wave][m][i];
            const float am  = al[wave][m];
            const float e0 = c1 * xmi * xm0;
            const float e1 = c1 * xmi * xm1;
            const float e2 = c1 * xmi * xmi;   // equals e0/e1 when i<2, safe in max
            const float mx = fmaxf(fmaxf(e0, e1), e2);
            const float w0 = __expf(e0 - mx);
            const float w1 = __expf(e1 - mx);
            const float w2 = g * __expf(e2 - mx);
            const float inv = __builtin_amdgcn_rcpf(w0 + w1 + w2);
            const float s  = (w0 * xm0 + w1 * xm1 + w2 * xmi) * inv;
            out[(s0 + m) * 16 + i] = acc[rr] + c2 * s * am;
        }
    }
}

extern "C" void kernel_launch(void* const* d_in, const int* in_sizes, int n_in,
                              void* d_out, int out_size, void* d_ws, size_t ws_size,
                              hipStream_t stream) {
    const float* x  = (const float*)d_in[0];
    const float* t  = (const float*)d_in[1];
    const float* Lb = (const float*)d_in[2];
    const float* wq = (const float*)d_in[3];
    const float* wk = (const float*)d_in[4];
    const float* wv = (const float*)d_in[5];
    const float* wo = (const float*)d_in[6];
    const float* aw = (const float*)d_in[7];
    float* out = (float*)d_out;

    const int B = in_sizes[1];        // t_years element count
    const int nTiles = B >> 4;        // 16 samples per tile
    if (nTiles <= 0) return;

    int blocks = (nTiles + WPB - 1) / WPB;
    if (blocks > 512) blocks = 512;   // grid-stride: ~2 tiles per wave at B=131072

    nxro_fused_kernel<<<blocks, 256, 0, stream>>>(x, t, Lb, wq, wk, wv, wo, aw,
                                                  out, nTiles);
}